// CDLoss_65987877536030
// MI455X (gfx1250) — compile-verified
//
#include <hip/hip_runtime.h>
#include <hip/hip_bf16.h>

typedef __attribute__((ext_vector_type(2))) float v2f;
typedef __attribute__((ext_vector_type(8))) float v8f;

#define MAXG 16

// ---------------- workspace layout (ints) ----------------
// wsI[0..15]   : counts[16]
// wsI[16..31]  : starts[16]
// wsI[32]      : max graph id
// wsI[36..39]  : meta = {ngraphs, n_max, tiles_per_graph, total_tiles}
// (float*)(wsI+48): sums[2]  (byte offset 192)

__global__ void cd_zero_kernel(int* wsI) {
    int i = threadIdx.x;
    if (i < MAXG) wsI[i] = 0;          // counts
    if (i == 0) {
        wsI[32] = -1;                  // max graph id
        float* sums = (float*)(wsI + 48);
        sums[0] = 0.0f;
        sums[1] = 0.0f;
    }
}

__global__ void cd_hist_kernel(const int* __restrict__ batch, int n, int* wsI) {
    int i = blockIdx.x * blockDim.x + threadIdx.x;
    if (i < n) {
        int b = batch[i];
        atomicAdd(&wsI[b], 1);
        atomicMax(&wsI[32], b);
    }
}

__global__ void cd_scan_kernel(int* wsI) {
    if (threadIdx.x == 0) {
        int ng = wsI[32] + 1;
        if (ng < 1) ng = 1;
        if (ng > MAXG) ng = MAXG;
        int acc = 0, nmax = 0;
        for (int g = 0; g < ng; ++g) {
            wsI[16 + g] = acc;         // starts
            int c = wsI[g];
            acc += c;
            if (c > nmax) nmax = c;
        }
        int tpg = (nmax + 15) >> 4;
        wsI[36] = ng;
        wsI[37] = nmax;
        wsI[38] = tpg;
        wsI[39] = ng * tpg;
    }
}

// One wave (32 lanes) processes one 16-row query tile vs all reference tiles
// of the same graph. Grid-stride over tiles; 8 waves per block. All loads are
// branchless (clamped address + cndmask) so EXEC stays full for WMMA and the
// compiler can pipeline global_load_b96 across unrolled iterations.
__global__ __launch_bounds__(256) void cd_chamfer_pass(
    const float* __restrict__ Q,   // query point cloud [N,3]
    const float* __restrict__ R,   // reference point cloud [N,3]
    const int* __restrict__ wsI,
    int npts,
    float* __restrict__ sum_out)   // accumulator for sum of per-row mins
{
    const int n_max = wsI[37];
    const int tpg   = wsI[38];
    const int total = wsI[39];

    const int lane = threadIdx.x & 31;
    const int wid  = (blockIdx.x * blockDim.x + threadIdx.x) >> 5;
    const int nwaves = (gridDim.x * blockDim.x) >> 5;
    const int half = lane >> 4;        // 0: K=0,1 half   1: K=2,3 half
    const int l15  = lane & 15;

    for (int t = wid; t < total; t += nwaves) {
        const int b  = t / tpg;
        const int xt = t - b * tpg;
        const int xrow0 = xt << 4;
        const int cnt = wsI[b];
        const int st  = wsI[16 + b];
        int cm1 = cnt - 1;
        if (cm1 < 0) cm1 = 0;
        int gmax = npts - 1;
        if (gmax < 0) gmax = 0;

        // ---- load my query row (dense row xr of graph b), branchless ----
        const int xr = xrow0 + l15;
        {
            // (address always in-bounds; value zeroed when row is padding)
        }
        int xrc = xr < cm1 ? xr : cm1;
        int gx = st + xrc; gx = gx < gmax ? gx : gmax; gx *= 3;
        float px = Q[gx + 0], py = Q[gx + 1], pz = Q[gx + 2];
        const bool xreal = (xr < cnt);
        px = xreal ? px : 0.0f;
        py = xreal ? py : 0.0f;
        pz = xreal ? pz : 0.0f;
        const float x2 = px * px + py * py + pz * pz;

        // A-matrix (16x4 f32): lanes 0-15 -> (K0,K1)=(x,y); lanes 16-31 -> (K2,K3)=(z,0)
        v2f a;
        a.x = half ? pz : px;
        a.y = half ? 0.0f : py;

        // broadcast x2 of the 8 rows this lane's D-VGPRs cover (M = half*8 + v)
        float ax2[8];
#pragma unroll
        for (int v = 0; v < 8; ++v)
            ax2[v] = __shfl(x2, (half << 3) + v, 32);   // ds_bpermute_b32

        float mn[8];
#pragma unroll
        for (int v = 0; v < 8; ++v) mn[v] = 3.0e38f;

        // ---- sweep all reference tiles of graph b ----
        for (int yt = 0; yt < tpg; ++yt) {
            const int yr = (yt << 4) + l15;
            int yrc = yr < cm1 ? yr : cm1;
            int g = st + yrc; g = g < gmax ? g : gmax; g *= 3;
            float qx = R[g + 0], qy = R[g + 1], qz = R[g + 2];
            const bool yreal = (yr < cnt);
            qx = yreal ? qx : 0.0f;
            qy = yreal ? qy : 0.0f;
            qz = yreal ? qz : 0.0f;
            float y2 = qx * qx + qy * qy + qz * qz;  // in-dense padding: real origin point
            y2 = (yr < n_max) ? y2 : 1.0e30f;        // outside dense matrix: never wins min

            // B-matrix (4x16 f32): lanes 0-15 -> (K0,K1); lanes 16-31 -> (K2,K3)
            v2f bb;
            bb.x = half ? qz : qx;
            bb.y = half ? 0.0f : qy;

            v8f c = {};
            c = __builtin_amdgcn_wmma_f32_16x16x4_f32(
                    false, a, false, bb, (short)0, c, false, false);

            // D layout: VGPR v = row M = half*8+v, col N = l15
#pragma unroll
            for (int v = 0; v < 8; ++v) {
                const float d = ax2[v] + y2 - 2.0f * c[v];
                mn[v] = fminf(mn[v], d);
            }
        }

        // ---- min over the 16 columns held by this half's lanes ----
#pragma unroll
        for (int v = 0; v < 8; ++v) {
            float m = mn[v];
            m = fminf(m, __shfl_xor(m, 1, 32));
            m = fminf(m, __shfl_xor(m, 2, 32));
            m = fminf(m, __shfl_xor(m, 4, 32));
            m = fminf(m, __shfl_xor(m, 8, 32));
            mn[v] = m;
        }

        // lane 0 of each half owns rows half*8 .. half*8+7
        if (l15 == 0) {
            float s = 0.0f;
#pragma unroll
            for (int v = 0; v < 8; ++v) {
                const int grow = xrow0 + (half << 3) + v;
                s += (grow < n_max) ? mn[v] : 0.0f;
            }
            atomicAdd(sum_out, s);
        }
    }
}

__global__ void cd_finalize_kernel(const int* wsI, float* out) {
    if (threadIdx.x == 0) {
        const float* sums = (const float*)(wsI + 48);
        float denom = (float)wsI[36] * (float)wsI[37];
        if (denom < 1.0f) denom = 1.0f;
        out[0] = sums[0] / denom + sums[1] / denom;
    }
}

extern "C" void kernel_launch(void* const* d_in, const int* in_sizes, int n_in,
                              void* d_out, int out_size, void* d_ws, size_t ws_size,
                              hipStream_t stream) {
    const float* pred   = (const float*)d_in[0];
    const float* target = (const float*)d_in[1];
    const int*   batch  = (const int*)d_in[2];
    const int n = in_sizes[2];            // number of points

    int* wsI = (int*)d_ws;
    float* sums = (float*)(wsI + 48);
    float* out = (float*)d_out;

    cd_zero_kernel<<<1, 64, 0, stream>>>(wsI);
    cd_hist_kernel<<<(n + 255) / 256, 256, 0, stream>>>(batch, n, wsI);
    cd_scan_kernel<<<1, 32, 0, stream>>>(wsI);

    // two symmetric passes: cham_x (pred->target) and cham_y (target->pred)
    cd_chamfer_pass<<<512, 256, 0, stream>>>(pred, target, wsI, n, &sums[0]);
    cd_chamfer_pass<<<512, 256, 0, stream>>>(target, pred, wsI, n, &sums[1]);

    cd_finalize_kernel<<<1, 32, 0, stream>>>(wsI, out);
}